// SAGEConv_11656541241918
// MI455X (gfx1250) — compile-verified
//
#include <hip/hip_runtime.h>

#define IN_F   128
#define OUT_F  256
#define KDIM   256           // 2*IN_F
#define BM     64            // nodes per block (4 M-tiles of 16)
#define LN_EPS 1e-5f

typedef float v2f __attribute__((ext_vector_type(2)));
typedef float v8f __attribute__((ext_vector_type(8)));

// ---------------------------------------------------------------------------
// Kernel 1: zero the workspace accumulators (neigh_sum [N*128] + deg [N])
// ---------------------------------------------------------------------------
__global__ void sage_zero_ws(float4* __restrict__ ws, int n4) {
    int i = blockIdx.x * blockDim.x + threadIdx.x;
    if (i < n4) ws[i] = make_float4(0.f, 0.f, 0.f, 0.f);
}

// ---------------------------------------------------------------------------
// Kernel 2: edge scatter. One wave32 per edge; lane L moves float4 at col 4L.
// feat (51 MB) is L2-resident (192 MB L2) so the random gather stays on-chip.
// ---------------------------------------------------------------------------
__global__ __launch_bounds__(256)
void sage_scatter(const float* __restrict__ feat,
                  const int*   __restrict__ esrc,
                  const int*   __restrict__ edst,
                  float*       __restrict__ nsum,
                  float*       __restrict__ deg,
                  int n_edges) {
    int wid  = (blockIdx.x * blockDim.x + threadIdx.x) >> 5;
    int lane = threadIdx.x & 31;
    if (wid >= n_edges) return;
    int s = esrc[wid];
    int d = edst[wid];
    float4 v = ((const float4*)(feat + (size_t)s * IN_F))[lane];
    float* dst = nsum + (size_t)d * IN_F + lane * 4;
    unsafeAtomicAdd(dst + 0, v.x);
    unsafeAtomicAdd(dst + 1, v.y);
    unsafeAtomicAdd(dst + 2, v.z);
    unsafeAtomicAdd(dst + 3, v.w);
    if (lane == 0) unsafeAtomicAdd(deg + d, 1.0f);
}

// ---------------------------------------------------------------------------
// Kernel 3: 64-node tile: stage concat rows in LDS, fp32 WMMA GEMM with
// 4 M-tiles x 2 N-tiles per wave (8 accumulators -> each B fragment feeds
// 4 v_wmma_f32_16x16x4_f32), bias+ReLU, LayerNorm. H aliases the X buffer.
// Block = 256 threads = 8 waves; wave w owns output columns [32w, 32w+32).
// ---------------------------------------------------------------------------
__global__ __launch_bounds__(256)
void sage_apply(const float* __restrict__ feat,
                const float* __restrict__ W,      // [256, 256] row-major (k, n)
                const float* __restrict__ bias,
                const float* __restrict__ gamma,
                const float* __restrict__ beta,
                const float* __restrict__ nsum,
                const float* __restrict__ deg,
                float*       __restrict__ out,
                int n_nodes) {
    __shared__ float Xs[BM][KDIM + 4];   // 66.5 KB; +4 pad: conflict-free cols
    float (*Hs)[KDIM + 4] = Xs;          // H aliases X (same shape), barriered

    const int tid   = threadIdx.x;
    const int node0 = blockIdx.x * BM;

    // ---- stage concat(feat, neigh_sum/deg) rows into LDS (16 float4/thread) ----
    #pragma unroll
    for (int i = 0; i < 16; ++i) {
        int f4   = tid + i * 256;          // 0..4095, 64 float4 per row
        int row  = f4 >> 6;
        int c4   = f4 & 63;
        int col  = c4 * 4;
        int node = node0 + row;
        float4 v = make_float4(0.f, 0.f, 0.f, 0.f);
        if (node < n_nodes) {
            if (col < IN_F) {
                v = ((const float4*)(feat + (size_t)node * IN_F))[c4];
            } else {
                float sc = 1.0f / fmaxf(deg[node], 1.0f);
                float4 t = ((const float4*)(nsum + (size_t)node * IN_F))[c4 - 32];
                v = make_float4(t.x * sc, t.y * sc, t.z * sc, t.w * sc);
            }
        }
        *(float4*)&Xs[row][col] = v;
    }
    __syncthreads();

    // ---- WMMA GEMM: 4 M-tiles x 2 N-tiles per wave, fp32 16x16x4 ----
    const int wave = tid >> 5;
    const int lane = tid & 31;
    const int lm   = lane & 15;
    const int kOff = (lane >> 4) << 1;     // lanes 0-15: K,K+1 ; lanes 16-31: K+2,K+3
    const int n0   = wave * 32;

    v8f c[4][2] = {};
    #pragma unroll 2
    for (int k = 0; k < KDIM; k += 4) {
        const float* wp = W + (size_t)(k + kOff) * OUT_F + n0 + lm;
        v2f b0, b1;
        b0.x = wp[0];
        b0.y = wp[OUT_F];
        b1.x = wp[16];
        b1.y = wp[OUT_F + 16];
        #pragma unroll
        for (int mt = 0; mt < 4; ++mt) {
            v2f a;
            a.x = Xs[mt * 16 + lm][k + kOff];
            a.y = Xs[mt * 16 + lm][k + kOff + 1];
            c[mt][0] = __builtin_amdgcn_wmma_f32_16x16x4_f32(false, a, false, b0,
                                                             (short)0, c[mt][0],
                                                             false, false);
            c[mt][1] = __builtin_amdgcn_wmma_f32_16x16x4_f32(false, a, false, b1,
                                                             (short)0, c[mt][1],
                                                             false, false);
        }
    }
    __syncthreads();   // all waves done reading Xs; safe to overwrite with H

    // ---- bias + ReLU, spill D tiles to LDS (D: VGPR r -> row r / r+8) ----
    const int   mAdd  = (lane >> 4) * 8;
    const float bias0 = bias[n0 + lm];
    const float bias1 = bias[n0 + 16 + lm];
    #pragma unroll
    for (int mt = 0; mt < 4; ++mt) {
        #pragma unroll
        for (int r = 0; r < 8; ++r) {
            int m = mt * 16 + r + mAdd;
            Hs[m][n0 + lm]      = fmaxf(c[mt][0][r] + bias0, 0.0f);
            Hs[m][n0 + 16 + lm] = fmaxf(c[mt][1][r] + bias1, 0.0f);
        }
    }
    __syncthreads();

    // ---- LayerNorm: 16 threads per node, shfl_xor within 16-lane groups ----
    const int l16 = tid & 15;
    #pragma unroll
    for (int nb = 0; nb < BM / 16; ++nb) {
        const int node = nb * 16 + (tid >> 4);
        if (node0 + node >= n_nodes) continue;

        float vals[16];
        float s = 0.f, ss = 0.f;
        #pragma unroll
        for (int j = 0; j < 16; ++j) {
            float h = Hs[node][l16 + j * 16];
            vals[j] = h;
            s  += h;
            ss += h * h;
        }
        #pragma unroll
        for (int m = 1; m < 16; m <<= 1) {
            s  += __shfl_xor(s,  m, 32);
            ss += __shfl_xor(ss, m, 32);
        }
        const float mu   = s * (1.0f / OUT_F);
        const float var  = ss * (1.0f / OUT_F) - mu * mu;
        const float rinv = rsqrtf(var + LN_EPS);

        float* op = out + (size_t)(node0 + node) * OUT_F;
        #pragma unroll
        for (int j = 0; j < 16; ++j) {
            int col = l16 + j * 16;
            op[col] = (vals[j] - mu) * rinv * gamma[col] + beta[col];
        }
    }
}

// ---------------------------------------------------------------------------
extern "C" void kernel_launch(void* const* d_in, const int* in_sizes, int n_in,
                              void* d_out, int out_size, void* d_ws, size_t ws_size,
                              hipStream_t stream) {
    const float* feat  = (const float*)d_in[0];
    const float* W     = (const float*)d_in[1];
    const float* bias  = (const float*)d_in[2];
    const float* gamma = (const float*)d_in[3];
    const float* beta  = (const float*)d_in[4];
    const int*   esrc  = (const int*)d_in[5];
    const int*   edst  = (const int*)d_in[6];

    const int n_nodes = in_sizes[0] / IN_F;
    const int n_edges = in_sizes[5];

    float* nsum = (float*)d_ws;                       // N * 128 floats
    float* deg  = nsum + (size_t)n_nodes * IN_F;      // N floats

    // 1) zero accumulators (harness poisons ws; must re-zero every call)
    {
        int n4  = (n_nodes * IN_F + n_nodes + 3) / 4;
        int blk = 256;
        sage_zero_ws<<<(n4 + blk - 1) / blk, blk, 0, stream>>>((float4*)d_ws, n4);
    }
    // 2) edge scatter: one wave per edge
    {
        int waves_per_block = 8;                      // 256 threads
        int blocks = (n_edges + waves_per_block - 1) / waves_per_block;
        sage_scatter<<<blocks, 256, 0, stream>>>(feat, esrc, edst, nsum, deg, n_edges);
    }
    // 3) fused concat + WMMA GEMM + bias + ReLU + LayerNorm
    {
        int blocks = (n_nodes + BM - 1) / BM;
        sage_apply<<<blocks, 256, 0, stream>>>(feat, W, bias, gamma, beta,
                                               nsum, deg, (float*)d_out, n_nodes);
    }
}